// EMHSLayer_51814485459480
// MI455X (gfx1250) — compile-verified
//
#include <hip/hip_runtime.h>
#include <hip/hip_bf16.h>

typedef __attribute__((ext_vector_type(2))) float v2f;
typedef __attribute__((ext_vector_type(8))) float v8f;

#define NVOX 23328   // B(32) * 729
#define SPB  729     // 9*9*9 spatial per batch
#define NV4  (NVOX / 4)

// ---------------- K0: zero scatter accumulators ----------------
__global__ void k_zero(float* __restrict__ p, int n) {
  int i = blockIdx.x * 256 + threadIdx.x;
  if (i < n) p[i] = 0.f;
}

// ---------------- K1: scatter-add x + counts per voxel ----------------
__global__ void k_scatter(const float* __restrict__ x, const int* __restrict__ vox,
                          float* __restrict__ gsum, float* __restrict__ cnt, int n) {
  int p = blockIdx.x * 256 + threadIdx.x;
  if (p >= n) return;
  int v = vox[p];
  atomicAdd(&gsum[v],            x[p * 3 + 0]);
  atomicAdd(&gsum[NVOX + v],     x[p * 3 + 1]);
  atomicAdd(&gsum[2 * NVOX + v], x[p * 3 + 2]);
  atomicAdd(&cnt[v], 1.0f);
}

// ---------------- K2: mean (empty voxels stay 0, as in reference), b128 ----------------
__global__ void k_mean(const float* __restrict__ gsum, const float* __restrict__ cnt,
                       float* __restrict__ g0) {
  int i4 = blockIdx.x * 256 + threadIdx.x;
  if (i4 >= 3 * NV4) return;
  int vr = i4 % NV4;
  float4 c = ((const float4*)cnt)[vr];
  float4 s = ((const float4*)gsum)[i4];
  float4 r;
  r.x = (c.x > 0.f) ? s.x / c.x : 0.f;
  r.y = (c.y > 0.f) ? s.y / c.y : 0.f;
  r.z = (c.z > 0.f) ? s.z / c.z : 0.f;
  r.w = (c.w > 0.f) ? s.w / c.w : 0.f;
  ((float4*)g0)[i4] = r;
}

// ---------------- K3: conv0 (3->64 ch, 3x3x3, pad 1). Tiny: plain VALU + LDS ----------------
// g0 viewed as (32, 3, 9,9,9); out0 = (32, 64, 9,9,9) flat.
__global__ void k_conv0(const float* __restrict__ g0, const float* __restrict__ w,
                        const float* __restrict__ bias, float* __restrict__ out0) {
  __shared__ float sh[81];
  int bs = blockIdx.x;                // 0..23327
  int b = bs / SPB, s = bs % SPB;
  int z = s / 81, y = (s / 9) % 9, xx = s % 9;
  int t = threadIdx.x;
  if (t < 81) {
    int c = t / 27, k = t % 27;
    int dz = k / 9 - 1, dy = (k / 3) % 3 - 1, dx = k % 3 - 1;
    int nz = z + dz, ny = y + dy, nx = xx + dx;
    bool val = (unsigned)nz < 9u && (unsigned)ny < 9u && (unsigned)nx < 9u;
    sh[t] = val ? g0[b * 2187 + c * 729 + nz * 81 + ny * 9 + nx] : 0.f;
  }
  __syncthreads();
  if (t < 64) {
    float acc = bias[t];
    const float* wr = w + t * 81;     // conv0_w: (64,3,3,3,3) -> o*81 + c*27 + k
    #pragma unroll
    for (int i = 0; i < 81; ++i) acc += sh[i] * wr[i];
    out0[b * 46656 + t * 729 + s] = acc;
  }
}

// ---------------- K4: build layer-1 conv input g1 in voxel space (b128) ----------------
// g1[j*NVOX + v] = occupied(v) ? mean_x[v]@W0[:,j] + b0[j] + out0colview[j,v] : 0
__global__ void k_g1(const float* __restrict__ g0, const float* __restrict__ cnt,
                     const float* __restrict__ w0, const float* __restrict__ b0,
                     const float* __restrict__ out0, float* __restrict__ g1) {
  int i4 = blockIdx.x * 256 + threadIdx.x;
  if (i4 >= 16 * NVOX) return;        // 64*NVOX / 4
  int j  = i4 / NV4;
  int vr = i4 % NV4;
  float bj  = b0[j];
  float w0j = w0[j], w1j = w0[64 + j], w2j = w0[128 + j]; // lin0_w: (3,64)
  float4 c4 = ((const float4*)cnt)[vr];
  float4 m0 = ((const float4*)g0)[vr];
  float4 m1 = ((const float4*)(g0 + NVOX))[vr];
  float4 m2 = ((const float4*)(g0 + 2 * NVOX))[vr];
  float4 o4 = ((const float4*)out0)[i4];                  // raw view (64, NVOX)
  float4 r;
  r.x = (c4.x > 0.f) ? (bj + m0.x * w0j + m1.x * w1j + m2.x * w2j + o4.x) : 0.f;
  r.y = (c4.y > 0.f) ? (bj + m0.y * w0j + m1.y * w1j + m2.y * w2j + o4.y) : 0.f;
  r.z = (c4.z > 0.f) ? (bj + m0.z * w0j + m1.z * w1j + m2.z * w2j + o4.z) : 0.f;
  r.w = (c4.w > 0.f) ? (bj + m0.w * w0j + m1.w * w1j + m2.w * w2j + o4.w) : 0.f;
  ((float4*)g1)[i4] = r;
}

// ---------------- K5: conv1 (64->64 ch) via V_WMMA_F32_16X16X4_F32 ----------------
// One wave per (16 spatial, 16 out-ch) tile; 27 taps x 16 K-chunks = 432 WMMAs.
__global__ void __launch_bounds__(32)
k_conv1_wmma(const float* __restrict__ g1, const float* __restrict__ w,
             const float* __restrict__ bias, float* __restrict__ out1) {
  int lane  = threadIdx.x;            // 0..31 (wave32; EXEC all ones for WMMA)
  int stile = blockIdx.x;             // 0..45 (46*16 = 736 >= 729)
  int otile = blockIdx.y;             // 0..3
  int b     = blockIdx.z;             // 0..31
  int mrow = lane & 15;
  int half = lane >> 4;
  int ksub = half * 2;                // A/B VGPR half-split: K+0/1 vs K+2/3
  int s = stile * 16 + mrow;
  int z = s / 81, y = (s / 9) % 9, xx = s % 9;
  int o = otile * 16 + mrow;
  const float* gb = g1 + (size_t)b * 46656;
  v8f acc = {};
  for (int k27 = 0; k27 < 27; ++k27) {
    int dz = k27 / 9 - 1, dy = (k27 / 3) % 3 - 1, dx = k27 % 3 - 1;
    int nz = z + dz, ny = y + dy, nx = xx + dx;
    bool val = (s < SPB) && (unsigned)nz < 9u && (unsigned)ny < 9u && (unsigned)nx < 9u;
    int ns = val ? (nz * 81 + ny * 9 + nx) : 0;      // clamp address, mask value
    float msk = val ? 1.f : 0.f;
    const float* ap = gb + ns + (size_t)ksub * 729;               // A: in[j, ns], j-stride 729
    const float* bp = w + (size_t)o * 1728 + (size_t)ksub * 27 + k27; // conv1_w (64,64,27)
    #pragma unroll
    for (int kc = 0; kc < 16; ++kc) {
      v2f a, bm;
      a.x = msk * ap[0];
      a.y = msk * ap[729];
      bm.x = bp[0];
      bm.y = bp[27];
      acc = __builtin_amdgcn_wmma_f32_16x16x4_f32(
          /*neg_a=*/false, a, /*neg_b=*/false, bm,
          /*c_mod=*/(short)0, acc, /*reuse_a=*/false, /*reuse_b=*/false);
      ap += 4 * 729;
      bp += 4 * 27;
    }
  }
  float bo = bias[o];
  #pragma unroll
  for (int r = 0; r < 8; ++r) {
    int sm = stile * 16 + r + half * 8;              // D: VGPR r -> M = r + 8*half
    if (sm < SPB)
      out1[(size_t)b * 46656 + (size_t)o * 729 + sm] = acc[r] + bo;
  }
}

// ---------------- K5b: fold weights: Weff = W0@W1 + Wf ; cb = b0@W1 + b1 + bf ----------------
__global__ void k_weff(const float* __restrict__ w0, const float* __restrict__ b0,
                       const float* __restrict__ w1, const float* __restrict__ b1,
                       const float* __restrict__ wf, const float* __restrict__ bf,
                       float* __restrict__ weff, float* __restrict__ cb) {
  int t = threadIdx.x;                 // 256 threads
  if (t < 192) {
    int c = t / 64, o = t % 64;
    float a = wf[c * 64 + o];
    for (int j = 0; j < 64; ++j) a += w0[c * 64 + j] * w1[j * 64 + o];
    weff[c * 64 + o] = a;
  } else {
    int o = t - 192;
    float a = b1[o] + bf[o];
    for (int j = 0; j < 64; ++j) a += b0[j] * w1[j * 64 + o];
    cb[o] = a;
  }
}

// ---------------- K6: per-voxel table g[v,o] = out0col[:,v]@W1 + out1col[o,v] + cb[o] (WMMA) ----
__global__ void __launch_bounds__(32)
k_gtab_wmma(const float* __restrict__ out0, const float* __restrict__ w1,
            const float* __restrict__ out1, const float* __restrict__ cb,
            float* __restrict__ gt) {
  int lane  = threadIdx.x;
  int vtile = blockIdx.x;              // 0..1457  (1458*16 == NVOX exactly)
  int otile = blockIdx.y;              // 0..3
  int mrow = lane & 15, half = lane >> 4, ksub = half * 2;
  int v = vtile * 16 + mrow;
  int o = otile * 16 + mrow;
  const float* ap = out0 + (size_t)ksub * NVOX + v;   // A[m, j] = out0[j*NVOX + v]
  const float* bp = w1 + ksub * 64 + o;               // B[j, o] = lin1_w[j*64 + o]
  v8f acc = {};
  #pragma unroll
  for (int kc = 0; kc < 16; ++kc) {
    v2f a, bm;
    a.x = ap[0];  a.y = ap[NVOX];
    bm.x = bp[0]; bm.y = bp[64];
    acc = __builtin_amdgcn_wmma_f32_16x16x4_f32(false, a, false, bm,
                                                (short)0, acc, false, false);
    ap += (size_t)4 * NVOX;
    bp += 4 * 64;
  }
  float c = cb[o];
  #pragma unroll
  for (int r = 0; r < 8; ++r) {
    int vm = vtile * 16 + r + half * 8;
    gt[(size_t)vm * 64 + o] = acc[r] + out1[(size_t)o * NVOX + vm] + c;
  }
}

// ---------------- K7: per-point output: out[p,:] = x[p]@Weff + g[voxel[p],:] (b128) ----------------
// One thread per float4 chunk: wave = 2 points x 16 chunks, fully coalesced 512B stores.
__global__ void k_final(const float* __restrict__ x, const int* __restrict__ vox,
                        const float* __restrict__ weff, const float* __restrict__ gt,
                        float* __restrict__ out, int n) {
  long long tid = (long long)blockIdx.x * 256 + threadIdx.x;
  if (tid >= (long long)n * 16) return;
  int p = (int)(tid >> 4);
  int q = (int)(tid & 15);             // which float4 of the 64 outputs
  int v = vox[p];
  float x0 = x[p * 3], x1 = x[p * 3 + 1], x2 = x[p * 3 + 2];
  float4 w0 = ((const float4*)weff)[q];
  float4 w1 = ((const float4*)weff)[16 + q];
  float4 w2 = ((const float4*)weff)[32 + q];
  float4 g  = ((const float4*)gt)[(size_t)v * 16 + q];
  float4 r;
  r.x = x0 * w0.x + x1 * w1.x + x2 * w2.x + g.x;
  r.y = x0 * w0.y + x1 * w1.y + x2 * w2.y + g.y;
  r.z = x0 * w0.z + x1 * w1.z + x2 * w2.z + g.z;
  r.w = x0 * w0.w + x1 * w1.w + x2 * w2.w + g.w;
  ((float4*)out)[tid] = r;
}

extern "C" void kernel_launch(void* const* d_in, const int* in_sizes, int n_in,
                              void* d_out, int out_size, void* d_ws, size_t ws_size,
                              hipStream_t stream) {
  const float* x       = (const float*)d_in[0];
  const float* lin0_w  = (const float*)d_in[1];
  const float* lin0_b  = (const float*)d_in[2];
  const float* conv0_w = (const float*)d_in[3];
  const float* conv0_b = (const float*)d_in[4];
  const float* lin1_w  = (const float*)d_in[5];
  const float* lin1_b  = (const float*)d_in[6];
  const float* conv1_w = (const float*)d_in[7];
  const float* conv1_b = (const float*)d_in[8];
  const float* linf_w  = (const float*)d_in[9];
  const float* linf_b  = (const float*)d_in[10];
  const int*   vox     = (const int*)d_in[12];   // cluster_non_consecutive
  float* out = (float*)d_out;
  int n = in_sizes[0] / 3;                        // N points

  // workspace layout (floats); all voxel-space buffers are L2-resident (~25 MB)
  float* ws   = (float*)d_ws;
  float* gsum = ws;                   // 3*NVOX
  float* cnt  = gsum + 3 * NVOX;      // NVOX
  float* g0   = cnt  + NVOX;          // 3*NVOX
  float* out0 = g0   + 3 * NVOX;      // 64*NVOX
  float* g1   = out0 + 64 * NVOX;     // 64*NVOX
  float* out1 = g1   + 64 * NVOX;     // 64*NVOX
  float* gt   = out1 + 64 * NVOX;     // 64*NVOX
  float* weff = gt   + 64 * NVOX;     // 192
  float* cb   = weff + 192;           // 64

  k_zero<<<(4 * NVOX + 255) / 256, 256, 0, stream>>>(gsum, 4 * NVOX); // gsum+cnt contiguous
  k_scatter<<<(n + 255) / 256, 256, 0, stream>>>(x, vox, gsum, cnt, n);
  k_mean<<<(3 * NV4 + 255) / 256, 256, 0, stream>>>(gsum, cnt, g0);
  k_conv0<<<NVOX, 128, 0, stream>>>(g0, conv0_w, conv0_b, out0);
  k_g1<<<(16 * NVOX + 255) / 256, 256, 0, stream>>>(g0, cnt, lin0_w, lin0_b, out0, g1);
  k_conv1_wmma<<<dim3(46, 4, 32), 32, 0, stream>>>(g1, conv1_w, conv1_b, out1);
  k_weff<<<1, 256, 0, stream>>>(lin0_w, lin0_b, lin1_w, lin1_b, linf_w, linf_b, weff, cb);
  k_gtab_wmma<<<dim3(NVOX / 16, 4), 32, 0, stream>>>(out0, lin1_w, out1, cb, gt);
  long long tot = (long long)n * 16;
  k_final<<<(unsigned)((tot + 255) / 256), 256, 0, stream>>>(x, vox, weff, gt, out, n);
}